// DKM_param_7679401526054
// MI455X (gfx1250) — compile-verified
//
#include <hip/hip_runtime.h>

typedef float v2f __attribute__((ext_vector_type(2)));
typedef float v8f __attribute__((ext_vector_type(8)));

#define TEMP_INV   20.0f
#define THRESHOLD  1e-4f
#define EPS_F      1e-6f

// ---------------------------------------------------------------------------
// Kernel Z: zero a scratch region (acc / colsum / diffsum)
// ---------------------------------------------------------------------------
__global__ __launch_bounds__(256) void zero_kernel(float* __restrict__ p, int n) {
    int i = blockIdx.x * 256 + threadIdx.x;
    if (i < n) p[i] = 0.0f;
}

// ---------------------------------------------------------------------------
// Kernel A: out[m, l] = sum_e A[m*lda + e] * W1[l*1536 + eoff + e]  (+ bias[l])
//   L1 = 512 fixed, K = 768 fixed.
//   Block = 256 thr (8 waves) covering a 128(m) x 64(l) tile:
//     blockIdx.x = l-tile (l0 = x*64), all 8 waves share the same 64 B-columns
//     (identical B fragments -> L0 hits); blockIdx.y*8 + wave = m-tile.
//   Waves whose m-tile >= mTiles exit (no barriers in this kernel).
//   Each wave: 16(m) x 64(l) via 4 f32 WMMA accumulators, K-step 4.
// ---------------------------------------------------------------------------
__global__ __launch_bounds__(256) void gemm_xw_kernel(
    const float* __restrict__ A, const float* __restrict__ W1,
    const float* __restrict__ bias, float* __restrict__ out,
    int lda, int eoff, int mTiles)
{
    const int lane = threadIdx.x & 31;
    const int wave = threadIdx.x >> 5;
    const int lo = lane & 15;
    const int hi = lane >> 4;

    const int mtile = blockIdx.y * 8 + wave;
    if (mtile >= mTiles) return;
    const int m0 = mtile * 16;
    const int l0 = blockIdx.x * 64;

    v8f acc[4] = {v8f{}, v8f{}, v8f{}, v8f{}};

    const float* arow = A + (size_t)(m0 + lo) * lda + 2 * hi;
    const float* bp0 = W1 + (size_t)(l0 +  0 + lo) * 1536 + eoff + 2 * hi;
    const float* bp1 = W1 + (size_t)(l0 + 16 + lo) * 1536 + eoff + 2 * hi;
    const float* bp2 = W1 + (size_t)(l0 + 32 + lo) * 1536 + eoff + 2 * hi;
    const float* bp3 = W1 + (size_t)(l0 + 48 + lo) * 1536 + eoff + 2 * hi;

    for (int e = 0; e < 768; e += 4) {
        // A fragment 16x4 f32: lane<16 -> K=e..e+1, lane>=16 -> K=e+2..e+3
        v2f a   = *(const v2f*)(arow + e);
        v2f b0  = *(const v2f*)(bp0 + e);
        v2f b1  = *(const v2f*)(bp1 + e);
        v2f b2v = *(const v2f*)(bp2 + e);
        v2f b3  = *(const v2f*)(bp3 + e);
        acc[0] = __builtin_amdgcn_wmma_f32_16x16x4_f32(false, a, false, b0,  (short)0, acc[0], false, false);
        acc[1] = __builtin_amdgcn_wmma_f32_16x16x4_f32(false, a, false, b1,  (short)0, acc[1], false, false);
        acc[2] = __builtin_amdgcn_wmma_f32_16x16x4_f32(false, a, false, b2v, (short)0, acc[2], false, false);
        acc[3] = __builtin_amdgcn_wmma_f32_16x16x4_f32(false, a, false, b3,  (short)0, acc[3], false, false);
    }

    // D layout: VGPR r -> M = r + 8*hi, N = lo
    #pragma unroll
    for (int j = 0; j < 4; ++j) {
        int l = l0 + j * 16 + lo;
        float bv = bias ? bias[l] : 0.0f;
        #pragma unroll
        for (int r = 0; r < 8; ++r) {
            int m = m0 + r + 8 * hi;
            out[(size_t)m * 512 + l] = acc[j][r] + bv;
        }
    }
}

// ---------------------------------------------------------------------------
// Kernel B: dist + softmax + a + colsum.
//   gk (=fc+b1) lives in LDS padded to stride 513 (conflict-free lane-per-k),
//   fx row staged per wave in LDS for broadcast reads. Lane owns clusters
//   k=lane and k=lane+32; softmax over 64 via lane-local pair + shfl_xor.
//   Grid 256 blocks x 8 waves, 2 rows (n) per wave -> 4096 rows.
//   LDS ~150 KB -> 2 blocks/WGP (320 KB).
// ---------------------------------------------------------------------------
__global__ __launch_bounds__(256) void dist_softmax_kernel(
    const float* __restrict__ fx, const float* __restrict__ gk,
    const float* __restrict__ w2, const float* __restrict__ b2p,
    float* __restrict__ aOut, float* __restrict__ colsum)
{
    extern __shared__ float smem[];
    float* gks = smem;                                     // 64 * 513
    float* w2s = smem + 64 * 513;                          // 512
    float* fxs = w2s + 512 + (threadIdx.x >> 5) * 512;     // per-wave 512

    for (int idx = threadIdx.x; idx < 64 * 512; idx += 256) {
        int r = idx >> 9, c = idx & 511;
        gks[r * 513 + c] = gk[idx];
    }
    for (int idx = threadIdx.x; idx < 512; idx += 256) w2s[idx] = w2[idx];
    __syncthreads();

    const int lane = threadIdx.x & 31;
    const int wave = threadIdx.x >> 5;
    const float b2v = b2p[0];
    const float* gk0 = gks + lane * 513;
    const float* gk1 = gks + (lane + 32) * 513;

    float cs0 = 0.0f, cs1 = 0.0f;
    const int nStart = (blockIdx.x * 8 + wave) * 2;

    for (int i = 0; i < 2; ++i) {
        int n = nStart + i;
        const float* fxrow = fx + (size_t)n * 512;
        #pragma unroll
        for (int t = 0; t < 16; ++t) fxs[lane + 32 * t] = fxrow[lane + 32 * t];
        __syncthreads();

        float s0 = 0.0f, s1 = 0.0f;
        #pragma unroll 8
        for (int l = 0; l < 512; ++l) {
            float xv = fxs[l];          // broadcast, conflict-free
            float w  = w2s[l];          // broadcast
            float h0 = xv + gk0[l];     // stride-513 -> distinct banks
            float h1 = xv + gk1[l];
            s0 = __builtin_fmaf(fmaxf(h0, 0.0f), w, s0);
            s1 = __builtin_fmaf(fmaxf(h1, 0.0f), w, s1);
        }
        // logits = -tanh(s+b2)/TEMP ; softmax over 64 clusters
        float l0v = -tanhf(s0 + b2v) * TEMP_INV;
        float l1v = -tanhf(s1 + b2v) * TEMP_INV;
        float m = fmaxf(l0v, l1v);
        for (int off = 16; off >= 1; off >>= 1) m = fmaxf(m, __shfl_xor(m, off, 32));
        float e0 = __expf(l0v - m), e1 = __expf(l1v - m);
        float sum = e0 + e1;
        for (int off = 16; off >= 1; off >>= 1) sum += __shfl_xor(sum, off, 32);
        float a0 = e0 / sum, a1 = e1 / sum;
        aOut[(size_t)n * 64 + lane]      = a0;
        aOut[(size_t)n * 64 + lane + 32] = a1;
        cs0 += a0; cs1 += a1;
        __syncthreads();
    }
    atomicAdd(&colsum[lane],      cs0);
    atomicAdd(&colsum[lane + 32], cs1);
}

// ---------------------------------------------------------------------------
// Kernel C: accC[k, e] += sum_n a[n,k] * X[n,e]   (M=64, N=768, K=4096)
//   f32 WMMA, wave tile 16(k) x 64(e), split-K by 32 with f32 global atomics.
//   Grid: 192 blocks x 8 waves = 1536 waves = 4(kt) x 12(et) x 32(split).
// ---------------------------------------------------------------------------
__global__ __launch_bounds__(256) void atx_kernel(
    const float* __restrict__ aArr, const float* __restrict__ X,
    float* __restrict__ accC)
{
    const int lane = threadIdx.x & 31;
    const int wave = threadIdx.x >> 5;
    const int lo = lane & 15;
    const int hi = lane >> 4;

    int W  = blockIdx.x * 8 + wave;
    int kt = W & 3;
    int et = (W >> 2) % 12;
    int st = W / 48;               // 0..31
    int c  = kt * 16 + lo;         // cluster index (M dim)
    int e0 = et * 64;
    int nStart = st * 128;

    v8f acc[4] = {v8f{}, v8f{}, v8f{}, v8f{}};

    for (int nn = nStart; nn < nStart + 128; nn += 4) {
        int na = nn + 2 * hi;      // this lane's first K (=n) index
        v2f a = { aArr[(size_t)na * 64 + c], aArr[(size_t)(na + 1) * 64 + c] };
        #pragma unroll
        for (int j = 0; j < 4; ++j) {
            int e = e0 + j * 16 + lo;
            v2f b = { X[(size_t)na * 768 + e], X[(size_t)(na + 1) * 768 + e] };
            acc[j] = __builtin_amdgcn_wmma_f32_16x16x4_f32(false, a, false, b, (short)0, acc[j], false, false);
        }
    }

    #pragma unroll
    for (int j = 0; j < 4; ++j) {
        #pragma unroll
        for (int r = 0; r < 8; ++r) {
            int k = kt * 16 + r + 8 * hi;
            int e = e0 + j * 16 + lo;
            atomicAdd(&accC[(size_t)k * 768 + e], acc[j][r]);
        }
    }
}

// ---------------------------------------------------------------------------
// Kernel D1: C_new = accC / (colsum + eps); diffsum += |C_new - C_init|
// ---------------------------------------------------------------------------
__global__ __launch_bounds__(256) void cnew_kernel(
    const float* __restrict__ accC, const float* __restrict__ colsum,
    const float* __restrict__ Cinit, float* __restrict__ Cnew,
    float* __restrict__ diffsum)
{
    int idx = blockIdx.x * 256 + threadIdx.x;  // < 49152
    int k = idx / 768;
    float cn = accC[idx] / (colsum[k] + EPS_F);
    Cnew[idx] = cn;
    float d = fabsf(cn - Cinit[idx]);
    for (int off = 16; off >= 1; off >>= 1) d += __shfl_xor(d, off, 32);
    if ((threadIdx.x & 31) == 0) atomicAdd(diffsum, d);
}

// ---------------------------------------------------------------------------
// Kernel D2: select C_new vs C_init based on diff threshold
// ---------------------------------------------------------------------------
__global__ __launch_bounds__(256) void select_kernel(
    const float* __restrict__ Cnew, const float* __restrict__ Cinit,
    const float* __restrict__ diffsum, float* __restrict__ outC)
{
    int idx = blockIdx.x * 256 + threadIdx.x;
    float diff = diffsum[0];
    outC[idx] = (diff > THRESHOLD) ? Cnew[idx] : Cinit[idx];
}

// ---------------------------------------------------------------------------
extern "C" void kernel_launch(void* const* d_in, const int* in_sizes, int n_in,
                              void* d_out, int out_size, void* d_ws, size_t ws_size,
                              hipStream_t stream) {
    const float* X     = (const float*)d_in[0];  // [4096, 768]
    const float* Cinit = (const float*)d_in[1];  // [64, 768]
    const float* W1    = (const float*)d_in[2];  // [512, 1536]
    const float* b1    = (const float*)d_in[3];  // [512]
    const float* w2    = (const float*)d_in[4];  // [512]
    const float* b2    = (const float*)d_in[5];  // scalar

    float* outC = (float*)d_out;                 // [64, 768]
    float* aOut = outC + 64 * 768;               // [4096, 64]

    float* ws      = (float*)d_ws;
    float* fx      = ws;                         // 4096*512
    float* gk      = fx + 4096 * 512;            // 64*512   (fc + b1)
    float* accC    = gk + 64 * 512;              // 64*768
    float* colsum  = accC + 64 * 768;            // 64
    float* diffsum = colsum + 64;                // 1
    float* Cnew    = diffsum + 1;                // 64*768

    // zero acc / colsum / diffsum (contiguous)
    zero_kernel<<<193, 256, 0, stream>>>(accC, 64 * 768 + 64 + 1);

    // fx = X @ W1x^T   : 256 m-tiles -> grid (8 l-tiles, 32 m-chunks)
    {
        dim3 g(8, 32);
        gemm_xw_kernel<<<g, 256, 0, stream>>>(X, W1, nullptr, fx, 768, 0, 256);
    }
    // gk = C @ W1c^T + b1 : 4 m-tiles -> grid (8, 1), waves 4..7 idle
    {
        dim3 g(8, 1);
        gemm_xw_kernel<<<g, 256, 0, stream>>>(Cinit, W1, b1, gk, 768, 768, 4);
    }

    // dist + softmax + colsum
    size_t smemB = (size_t)(64 * 513 + 512 + 8 * 512) * sizeof(float);
    dist_softmax_kernel<<<256, 256, smemB, stream>>>(fx, gk, w2, b2, aOut, colsum);

    // accC = a^T @ X (split-K by 32, atomics)
    atx_kernel<<<192, 256, 0, stream>>>(aOut, X, accC);

    // finalize
    cnew_kernel<<<192, 256, 0, stream>>>(accC, colsum, Cinit, Cnew, diffsum);
    select_kernel<<<192, 256, 0, stream>>>(Cnew, Cinit, diffsum, outC);
}